// CapsNet_5411658793440
// MI455X (gfx1250) — compile-verified
//
#include <hip/hip_runtime.h>

#define NUM_CLASSES 14
#define PRIM_DIM 32
#define OUT_DIM 16
#define IMG 224
#define FEAT 108
#define S_SPAT (FEAT * FEAT)                 // 11664
#define NUM_ROUTES (8 * S_SPAT)              // 93312
#define BATCH 2
#define MTOT (BATCH * S_SPAT)                // 23328 (M of conv GEMM)
#define KTOT 243                             // 3*9*9
#define OC 256

#define NCHUNK 36                            // softmax-stats r-chunks per (b,c)
#define RCH (NUM_ROUTES / NCHUNK)            // 2592
#define SJ_CH 81                             // s_j / b-update r-chunks per (b,c)
#define SJ_R (NUM_ROUTES / SJ_CH)            // 1152

typedef float v2f __attribute__((ext_vector_type(2)));
typedef float v4f __attribute__((ext_vector_type(4)));
typedef float v8f __attribute__((ext_vector_type(8)));

// ---------- order-preserving float<->uint for atomicMax ----------
__device__ __forceinline__ unsigned fenc(float f) {
  unsigned b = __float_as_uint(f);
  return (b & 0x80000000u) ? ~b : (b | 0x80000000u);
}
__device__ __forceinline__ float fdec(unsigned e) {
  unsigned b = (e & 0x80000000u) ? (e ^ 0x80000000u) : ~e;
  return __uint_as_float(b);
}

// im2col flat-K -> x offset relative to (iy0,ix0) pixel base; constant-folds under unroll
__device__ __forceinline__ int im2col_off(int kk) {
  int ic = kk / 81, r2 = kk - ic * 81, kh = r2 / 9, kw = r2 - kh * 9;
  return ic * (IMG * IMG) + kh * IMG + kw;
}

// ---------- generic zero ----------
__global__ void zero_f32(float* __restrict__ p, long n) {
  long i = (long)blockIdx.x * blockDim.x + threadIdx.x;
  if (i < n) p[i] = 0.f;
}

// ---------- conv 9x9 stride2 as implicit-im2col GEMM via f32 WMMA ----------
// out y: [MTOT][OC] channel-last; one wave -> one 16x16 tile; K fully unrolled (61 WMMAs).
__global__ __launch_bounds__(256) void conv_wmma(const float* __restrict__ x,
                                                 const float* __restrict__ w,
                                                 const float* __restrict__ bias,
                                                 float* __restrict__ y) {
  const int wid   = (blockIdx.x * blockDim.x + threadIdx.x) >> 5;
  const int lane  = threadIdx.x & 31;
  const int mtile = wid >> 4;                 // 1458 M tiles
  const int ntile = wid & 15;                 // 16 N tiles
  const int half  = lane >> 4;
  const int row   = lane & 15;

  // A-row (im2col row) owned by this lane
  const int m  = mtile * 16 + row;
  const int b  = m / S_SPAT;
  const int rm = m - b * S_SPAT;
  const int oy = rm / FEAT;
  const int ox = rm - oy * FEAT;
  const float* xp = x + (size_t)b * 3 * IMG * IMG + (size_t)(oy * 2) * IMG + (ox * 2);

  // B-column (output channel) owned by this lane
  const int ncol = ntile * 16 + row;
  const float* wn = w + (size_t)ncol * KTOT;  // conv_w is [OC][3*9*9] contiguous

  v8f acc = {};
#pragma unroll
  for (int k = 0; k < KTOT; k += 4) {
    // lanes 0-15 supply K = k,k+1 (VGPR0/VGPR1); lanes 16-31 supply K = k+2,k+3
    const int kA    = half ? (k + 2) : k;                       // cndmask of constants
    const int offA0 = half ? im2col_off(k + 2) : im2col_off(k); // cndmask of constants
    float a0 = xp[offA0];
    float b0 = wn[kA];
    float a1 = 0.f, b1 = 0.f;
    if (kA + 1 < KTOT) {                      // divergent only in the final K chunk
      const int offA1 = half ? im2col_off(k + 3 < KTOT ? k + 3 : 0)
                             : im2col_off(k + 1);
      a1 = xp[offA1];
      b1 = wn[kA + 1];
    }
    v2f af = {a0, a1};
    v2f bf = {b0, b1};
    acc = __builtin_amdgcn_wmma_f32_16x16x4_f32(false, af, false, bf,
                                                (short)0, acc, false, false);
  }
  const float bv = bias[ncol];
#pragma unroll
  for (int vv = 0; vv < 8; ++vv) {            // C/D layout: VGPR v = rows v / v+8
    int orow = mtile * 16 + vv + half * 8;
    y[(size_t)orow * OC + ncol] = acc[vv] + bv;
  }
}

// ---------- primary-capsule squash: u[b][r][32], r = s*8 + p ----------
__global__ void squash_prim(const float* __restrict__ y, float* __restrict__ u) {
  int t = blockIdx.x * blockDim.x + threadIdx.x;   // B*S*8 = 186624 threads
  if (t >= BATCH * S_SPAT * 8) return;
  int p = t & 7, bs = t >> 3;
  const v4f* src = (const v4f*)(y + (size_t)bs * OC + p * PRIM_DIM);
  v4f vals[8];
  float n2 = 0.f;
#pragma unroll
  for (int i = 0; i < 8; ++i) {
    vals[i] = src[i];
    n2 += vals[i][0] * vals[i][0] + vals[i][1] * vals[i][1] +
          vals[i][2] * vals[i][2] + vals[i][3] * vals[i][3];
  }
  float n = sqrtf(n2);
  float f = n / ((1.f + n2) * (1.f + n2));         // (n^2/(1+n^2)^2) * x/n
  int bb = bs / S_SPAT, s = bs - bb * S_SPAT;
  v4f* dst = (v4f*)(u + ((size_t)bb * NUM_ROUTES + (size_t)s * 8 + p) * PRIM_DIM);
#pragma unroll
  for (int i = 0; i < 8; ++i) dst[i] = vals[i] * f;
}

// ---------- u_hat[b][r][c][16] = u[b][r][:] @ route_w[c][r][:][:] ----------
// One wave per (c,r). Lane l owns o-group og=l&3 (4 outputs) and i-slice l>>2;
// 4 x B128 loads sweep the 2KB route_w block (the 2.67GB stream).
__global__ __launch_bounds__(256) void uhat_kernel(const float* __restrict__ u,
                                                   const float* __restrict__ rw,
                                                   float* __restrict__ uh) {
  int wid  = (blockIdx.x * blockDim.x + threadIdx.x) >> 5;
  int lane = threadIdx.x & 31;
  int c = wid / NUM_ROUTES;
  int r = wid - c * NUM_ROUTES;
  float u0 = u[((size_t)0 * NUM_ROUTES + r) * PRIM_DIM + lane];
  float u1 = u[((size_t)1 * NUM_ROUTES + r) * PRIM_DIM + lane];
  const float* base = rw + ((size_t)c * NUM_ROUTES + r) * (PRIM_DIM * OUT_DIM);
  const int og = lane & 3, isl = lane >> 2;
  v4f acc0 = {}, acc1 = {};
#pragma unroll
  for (int it = 0; it < 4; ++it) {
    v4f w4 = *(const v4f*)(base + it * 128 + lane * 4);   // i = it*8+isl, o = og*4+j
    int i = it * 8 + isl;
    float a = __shfl(u0, i, 32);
    float bb = __shfl(u1, i, 32);
    acc0 += w4 * a;
    acc1 += w4 * bb;
  }
  // sum over the 8 lanes sharing og (i-slices)
#pragma unroll
  for (int off = 4; off <= 16; off <<= 1) {
#pragma unroll
    for (int j = 0; j < 4; ++j) {
      acc0[j] += __shfl_xor(acc0[j], off, 32);
      acc1[j] += __shfl_xor(acc1[j], off, 32);
    }
  }
  if (lane < 4) {                             // lane == og here
    v4f* d0 = (v4f*)(uh + (((size_t)0 * NUM_ROUTES + r) * NUM_CLASSES + c) * OUT_DIM + og * 4);
    v4f* d1 = (v4f*)(uh + (((size_t)1 * NUM_ROUTES + r) * NUM_CLASSES + c) * OUT_DIM + og * 4);
    *d0 = acc0;
    *d1 = acc1;
  }
}

// ---------- softmax stats over R (b_ij layout [B][C][R], r-contiguous) ----------
__global__ void init_stats(unsigned* __restrict__ mx, float* __restrict__ se) {
  int t = threadIdx.x;
  if (t < BATCH * NUM_CLASSES) { mx[t] = 0u; se[t] = 0.f; }   // 0u encodes < everything
}

__global__ void stats_max(const float* __restrict__ bij, unsigned* __restrict__ mx) {
  int bc = blockIdx.x / NCHUNK, ch = blockIdx.x % NCHUNK;
  const float* src = bij + (size_t)bc * NUM_ROUTES;
  float m = -__builtin_inff();
  for (int r = ch * RCH + threadIdx.x; r < (ch + 1) * RCH; r += blockDim.x)
    m = fmaxf(m, src[r]);
  for (int off = 16; off > 0; off >>= 1) m = fmaxf(m, __shfl_xor(m, off, 32));
  __shared__ float sm[8];
  if ((threadIdx.x & 31) == 0) sm[threadIdx.x >> 5] = m;
  __syncthreads();
  if (threadIdx.x == 0) {
    float mm = sm[0];
    for (int i = 1; i < (int)(blockDim.x >> 5); ++i) mm = fmaxf(mm, sm[i]);
    atomicMax(&mx[bc], fenc(mm));
  }
}

__global__ void stats_sum(const float* __restrict__ bij, const unsigned* __restrict__ mx,
                          float* __restrict__ se) {
  int bc = blockIdx.x / NCHUNK, ch = blockIdx.x % NCHUNK;
  const float* src = bij + (size_t)bc * NUM_ROUTES;
  float m = fdec(mx[bc]);
  float s = 0.f;
  for (int r = ch * RCH + threadIdx.x; r < (ch + 1) * RCH; r += blockDim.x)
    s += expf(src[r] - m);
  for (int off = 16; off > 0; off >>= 1) s += __shfl_xor(s, off, 32);
  __shared__ float sm[8];
  if ((threadIdx.x & 31) == 0) sm[threadIdx.x >> 5] = s;
  __syncthreads();
  if (threadIdx.x == 0) {
    float ss = 0.f;
    for (int i = 0; i < (int)(blockDim.x >> 5); ++i) ss += sm[i];
    atomicAdd(&se[bc], ss);
  }
}

// ---------- s_j[b][c][16] = sum_r c_ij * u_hat ----------
// Lane l: r-slot l>>2 (8 r per wave/step), o-group l&3; B128 loads of u_hat rows.
__global__ __launch_bounds__(256) void sj_kernel(const float* __restrict__ uh,
                                                 const float* __restrict__ bij,
                                                 const unsigned* __restrict__ mx,
                                                 const float* __restrict__ se,
                                                 float* __restrict__ sj, int uniform) {
  int bc = blockIdx.x / SJ_CH, ch = blockIdx.x % SJ_CH;
  int b = bc / NUM_CLASSES, c = bc % NUM_CLASSES;
  float m = 0.f, inv = 1.f / (float)NUM_ROUTES;
  if (!uniform) { m = fdec(mx[bc]); inv = 1.f / se[bc]; }
  int lane = threadIdx.x & 31, wv = threadIdx.x >> 5;
  int rslot = lane >> 2, og = lane & 3;
  v4f acc = {};
  int rend = ch * SJ_R + SJ_R;
  for (int r = ch * SJ_R + wv * 8 + rslot; r < rend; r += 64) {
    float cij = uniform ? inv : expf(bij[(size_t)bc * NUM_ROUTES + r] - m) * inv;
    v4f uv = *(const v4f*)(uh + (((size_t)b * NUM_ROUTES + r) * NUM_CLASSES + c) * OUT_DIM + og * 4);
    acc += uv * cij;
  }
#pragma unroll
  for (int off = 4; off <= 16; off <<= 1)
#pragma unroll
    for (int j = 0; j < 4; ++j) acc[j] += __shfl_xor(acc[j], off, 32);
  __shared__ float ssum[OUT_DIM];
  if (threadIdx.x < OUT_DIM) ssum[threadIdx.x] = 0.f;
  __syncthreads();
  if (lane < 4) {                             // lane == og
#pragma unroll
    for (int j = 0; j < 4; ++j) atomicAdd(&ssum[og * 4 + j], acc[j]);
  }
  __syncthreads();
  if (threadIdx.x < OUT_DIM) atomicAdd(&sj[bc * OUT_DIM + threadIdx.x], ssum[threadIdx.x]);
}

// ---------- v = squash(s_j), 28 vectors of 16 ----------
__global__ void squash_v(const float* __restrict__ sj, float* __restrict__ v) {
  int t = threadIdx.x;                        // blockDim = 448 = 28*16 (14 waves)
  float s = sj[t];
  float n2 = s * s;
  for (int off = 1; off < 16; off <<= 1) n2 += __shfl_xor(n2, off, 32);
  float n = sqrtf(n2);
  v[t] = s * (n / (1.f + n2));                // (n^2/(1+n^2)) * s/n
}

// ---------- b_ij[b][c][r] += sum_o v[b][c][o]*u_hat[b][r][c][o] ----------
__global__ __launch_bounds__(256) void bupd_kernel(const float* __restrict__ uh,
                                                   const float* __restrict__ v,
                                                   float* __restrict__ bij) {
  int bc = blockIdx.x / SJ_CH, ch = blockIdx.x % SJ_CH;
  int b = bc / NUM_CLASSES, c = bc % NUM_CLASSES;
  int lane = threadIdx.x & 31, wv = threadIdx.x >> 5;
  int rslot = lane >> 2, og = lane & 3;
  v4f vo = *(const v4f*)(v + bc * OUT_DIM + og * 4);
  int rend = ch * SJ_R + SJ_R;
  for (int r = ch * SJ_R + wv * 8 + rslot; r < rend; r += 64) {
    v4f uv = *(const v4f*)(uh + (((size_t)b * NUM_ROUTES + r) * NUM_CLASSES + c) * OUT_DIM + og * 4);
    float p = vo[0] * uv[0] + vo[1] * uv[1] + vo[2] * uv[2] + vo[3] * uv[3];
    p += __shfl_xor(p, 1, 32);                // fold the 4 o-groups (same r)
    p += __shfl_xor(p, 2, 32);
    if (og == 0) bij[(size_t)bc * NUM_ROUTES + r] += p;  // sole writer per (b,c,r)
  }
}

// ---------- final FC: out[b][j] = flat[b] . fc_w[j] + fc_b[j] ----------
__global__ void fc_kernel(const float* __restrict__ v, const float* __restrict__ fw,
                          const float* __restrict__ fb, float* __restrict__ out) {
  int t = threadIdx.x;
  if (t >= BATCH * NUM_CLASSES) return;
  int b = t / NUM_CLASSES, j = t % NUM_CLASSES;
  float acc = fb[j];
  const float* f = v + b * NUM_CLASSES * OUT_DIM;
  const float* w = fw + j * NUM_CLASSES * OUT_DIM;
  for (int k = 0; k < NUM_CLASSES * OUT_DIM; ++k) acc += f[k] * w[k];
  out[b * NUM_CLASSES + j] = acc;
}

extern "C" void kernel_launch(void* const* d_in, const int* in_sizes, int n_in,
                              void* d_out, int out_size, void* d_ws, size_t ws_size,
                              hipStream_t stream) {
  const float* x       = (const float*)d_in[0];
  const float* conv_w  = (const float*)d_in[1];
  const float* conv_b  = (const float*)d_in[2];
  const float* route_w = (const float*)d_in[3];
  const float* fc_w    = (const float*)d_in[4];
  const float* fc_b    = (const float*)d_in[5];
  float* out = (float*)d_out;

  // workspace carve-up (floats)
  const size_t NY  = (size_t)MTOT * OC;                                   // 5,971,968
  const size_t NU  = (size_t)BATCH * NUM_ROUTES * PRIM_DIM;               // 5,971,968
  const size_t NUH = (size_t)BATCH * NUM_ROUTES * NUM_CLASSES * OUT_DIM;  // 41,803,776
  const size_t NB  = (size_t)BATCH * NUM_CLASSES * NUM_ROUTES;            // 2,612,736
  float*    Y    = (float*)d_ws;
  float*    U    = Y + NY;
  float*    UH   = U + NU;
  float*    BIJ  = UH + NUH;
  float*    SJ   = BIJ + NB;
  float*    V    = SJ + BATCH * NUM_CLASSES * OUT_DIM;
  unsigned* MAXE = (unsigned*)(V + BATCH * NUM_CLASSES * OUT_DIM);
  float*    SUME = (float*)(MAXE + BATCH * NUM_CLASSES);

  // b_ij = 0
  zero_f32<<<(int)((NB + 255) / 256), 256, 0, stream>>>(BIJ, (long)NB);

  // conv (WMMA implicit GEMM): 1458*16 waves / 8 per block
  conv_wmma<<<(MTOT / 16) * 16 / 8, 256, 0, stream>>>(x, conv_w, conv_b, Y);

  // primary squash
  squash_prim<<<(BATCH * S_SPAT * 8) / 256, 256, 0, stream>>>(Y, U);

  // u_hat: one wave per (c,r)
  uhat_kernel<<<(NUM_CLASSES * NUM_ROUTES) / 8, 256, 0, stream>>>(U, route_w, UH);

  // dynamic routing, 3 iterations
  for (int it = 0; it < 3; ++it) {
    if (it > 0) {
      init_stats<<<1, 64, 0, stream>>>(MAXE, SUME);
      stats_max<<<BATCH * NUM_CLASSES * NCHUNK, 256, 0, stream>>>(BIJ, MAXE);
      stats_sum<<<BATCH * NUM_CLASSES * NCHUNK, 256, 0, stream>>>(BIJ, MAXE, SUME);
    }
    zero_f32<<<2, 256, 0, stream>>>(SJ, BATCH * NUM_CLASSES * OUT_DIM);
    sj_kernel<<<BATCH * NUM_CLASSES * SJ_CH, 256, 0, stream>>>(UH, BIJ, MAXE, SUME, SJ,
                                                               it == 0 ? 1 : 0);
    squash_v<<<1, BATCH * NUM_CLASSES * OUT_DIM, 0, stream>>>(SJ, V);
    if (it < 2)
      bupd_kernel<<<BATCH * NUM_CLASSES * SJ_CH, 256, 0, stream>>>(UH, V, BIJ);
  }

  fc_kernel<<<1, 32, 0, stream>>>(V, fc_w, fc_b, out);
}